// RecurrentPassband_22668837388340
// MI455X (gfx1250) — compile-verified
//
#include <hip/hip_runtime.h>

typedef __attribute__((ext_vector_type(2))) float v2f;
typedef __attribute__((ext_vector_type(8))) float v8f;

#define B_SZ 4096
#define T_SZ 256
#define DH   32
#define WAVES 4
#define ROWS_PER_WAVE 16

__device__ __forceinline__ float fsig(float x) {
    // 1/(1+exp(-x)) via v_exp_f32 (base-2) + v_rcp_f32
    return __builtin_amdgcn_rcpf(1.0f + __builtin_amdgcn_exp2f(-1.442695041f * x));
}

__global__ __launch_bounds__(WAVES * 32, 1)
void gru_wmma_kernel(const float* __restrict__ x,
                     const int*   __restrict__ lengths,
                     const float* __restrict__ W_ih,
                     const float* __restrict__ W_hh,
                     const float* __restrict__ b_ih,
                     const float* __restrict__ b_hh,
                     float* __restrict__ out)
{
    // per-wave private 16x36 A-staging tile: cols 0..31 = h, 32..34 = x_t, 35 = 1.0
    __shared__ float sh[WAVES][ROWS_PER_WAVE][36];

    const int tid  = threadIdx.x;
    const int lane = tid & 31;
    const int wv   = tid >> 5;
    const int nl   = lane & 15;   // N index within 16-wide tile / M row for A loads
    const int half = lane >> 4;   // 0: lanes 0-15, 1: lanes 16-31
    const int row0 = (blockIdx.x * WAVES + wv) * ROWS_PER_WAVE;

    // ---------------- B fragments, resident in VGPRs for whole kernel ----------------
    // B element (K,N): VGPR0 holds K = 4*kk + 2*half, VGPR1 holds K+1; N = nl + 16*tile
    v2f Brz[4][9];   // gates r,z: cols 0..63.  K rows: [W_hh | W_ih | b_ih+b_hh]
    v2f Bhn[2][9];   // h_n: cols 64..95.       K rows: [W_hh | 0    | b_hh]
    v2f Bin[2];      // i_n: cols 64..95, kk=8 only: rows [W_ih | b_ih]
#pragma unroll
    for (int nt = 0; nt < 4; ++nt) {
        const int g = nt * 16 + nl;          // gate column 0..63
#pragma unroll
        for (int kk = 0; kk < 8; ++kk) {
            const int K0 = 4 * kk + 2 * half;
            v2f f; f.x = W_hh[g * DH + K0]; f.y = W_hh[g * DH + K0 + 1];
            Brz[nt][kk] = f;
        }
        v2f f;
        if (half == 0) { f.x = W_ih[g * 3 + 0]; f.y = W_ih[g * 3 + 1]; }
        else           { f.x = W_ih[g * 3 + 2]; f.y = b_ih[g] + b_hh[g]; }
        Brz[nt][8] = f;
    }
#pragma unroll
    for (int nt = 0; nt < 2; ++nt) {
        const int g = 64 + nt * 16 + nl;     // gate column 64..95
#pragma unroll
        for (int kk = 0; kk < 8; ++kk) {
            const int K0 = 4 * kk + 2 * half;
            v2f f; f.x = W_hh[g * DH + K0]; f.y = W_hh[g * DH + K0 + 1];
            Bhn[nt][kk] = f;
        }
        v2f fh; fh.x = 0.0f; fh.y = half ? b_hh[g] : 0.0f;
        Bhn[nt][8] = fh;
        v2f fi;
        if (half == 0) { fi.x = W_ih[g * 3 + 0]; fi.y = W_ih[g * 3 + 1]; }
        else           { fi.x = W_ih[g * 3 + 2]; fi.y = b_ih[g]; }
        Bin[nt] = fi;
    }

    // lengths for the 8 rows whose C elements this lane holds (M = v + 8*half)
    int lenv[8];
#pragma unroll
    for (int v = 0; v < 8; ++v) lenv[v] = lengths[row0 + v + 8 * half];

    // h state in C/D layout: h[tt][v] = h[M = v+8*half][d = nl + 16*tt]
    float h[2][8];
#pragma unroll
    for (int tt = 0; tt < 2; ++tt)
#pragma unroll
        for (int v = 0; v < 8; ++v) h[tt][v] = 0.0f;

    // init LDS: h cols zero, col 35 = 1.0 (bias lane of augmented A)
    if (lane < 16) {
#pragma unroll
        for (int d = 0; d < 32; ++d) sh[wv][lane][d] = 0.0f;
        sh[wv][lane][35] = 1.0f;
    }

    const size_t hid_base = (size_t)B_SZ * T_SZ * DH;

    for (int t = 0; t < T_SZ; ++t) {
        // stage x_t into A cols 32..34
        if (lane < 16) {
            const float* xp = x + ((size_t)(row0 + lane) * T_SZ + t) * 3;
            sh[wv][lane][32] = xp[0];
            sh[wv][lane][33] = xp[1];
            sh[wv][lane][34] = xp[2];
        }
        asm volatile("s_wait_dscnt 0" ::: "memory");
        __builtin_amdgcn_wave_barrier();

        v8f Crz[4], Chn[2], Cin[2];
#pragma unroll
        for (int i = 0; i < 4; ++i) Crz[i] = {};
#pragma unroll
        for (int i = 0; i < 2; ++i) { Chn[i] = {}; Cin[i] = {}; }

        // D = [h | x | 1] (16x36) x B (36x96ish), 9 k-steps of 16x16x4 f32 WMMA
#pragma unroll
        for (int kk = 0; kk < 9; ++kk) {
            v2f a;
            const float* ap = &sh[wv][nl][4 * kk + 2 * half];  // ds_load_b64
            a.x = ap[0]; a.y = ap[1];
#pragma unroll
            for (int nt = 0; nt < 4; ++nt)
                Crz[nt] = __builtin_amdgcn_wmma_f32_16x16x4_f32(
                    false, a, false, Brz[nt][kk], (short)0, Crz[nt], false, false);
#pragma unroll
            for (int nt = 0; nt < 2; ++nt)
                Chn[nt] = __builtin_amdgcn_wmma_f32_16x16x4_f32(
                    false, a, false, Bhn[nt][kk], (short)0, Chn[nt], false, false);
            if (kk == 8) {
#pragma unroll
                for (int nt = 0; nt < 2; ++nt)
                    Cin[nt] = __builtin_amdgcn_wmma_f32_16x16x4_f32(
                        false, a, false, Bin[nt], (short)0, Cin[nt], false, false);
            }
        }

        // gates + state update + masked output, all in C layout
#pragma unroll
        for (int tt = 0; tt < 2; ++tt) {
#pragma unroll
            for (int v = 0; v < 8; ++v) {
                const int M = v + 8 * half;
                const int d = nl + 16 * tt;
                const float r = fsig(Crz[tt][v]);         // tiles 0,1 -> r
                const float z = fsig(Crz[2 + tt][v]);     // tiles 2,3 -> z
                const float nval =
                    2.0f * fsig(2.0f * (Cin[tt][v] + r * Chn[tt][v])) - 1.0f; // tanh
                const float hnew = (1.0f - z) * nval + z * h[tt][v];
                const bool  m    = (t < lenv[v]);
                const float hnx  = m ? hnew : h[tt][v];   // hidden freezes past length
                h[tt][v] = hnx;
                out[((size_t)(row0 + M) * T_SZ + t) * DH + d] = m ? hnew : 0.0f;
                sh[wv][M][d] = hnx;                       // h for next step's A
            }
        }
    }

    // final hidden state
#pragma unroll
    for (int tt = 0; tt < 2; ++tt)
#pragma unroll
        for (int v = 0; v < 8; ++v) {
            const int M = v + 8 * half;
            const int d = nl + 16 * tt;
            out[hid_base + (size_t)(row0 + M) * DH + d] = h[tt][v];
        }
}

extern "C" void kernel_launch(void* const* d_in, const int* in_sizes, int n_in,
                              void* d_out, int out_size, void* d_ws, size_t ws_size,
                              hipStream_t stream) {
    const float* x       = (const float*)d_in[0];
    const int*   lengths = (const int*)  d_in[1];
    const float* W_ih    = (const float*)d_in[2];
    const float* W_hh    = (const float*)d_in[3];
    const float* b_ih    = (const float*)d_in[4];
    const float* b_hh    = (const float*)d_in[5];
    float* out = (float*)d_out;

    dim3 grid(B_SZ / (WAVES * ROWS_PER_WAVE));   // 64 blocks
    dim3 block(WAVES * 32);                      // 4 waves of 32
    gru_wmma_kernel<<<grid, block, 0, stream>>>(x, lengths, W_ih, W_hh, b_ih, b_hh, out);
}